// TopKGate_37349035606099
// MI455X (gfx1250) — compile-verified
//
#include <hip/hip_runtime.h>

typedef __attribute__((ext_vector_type(2))) float v2f;
typedef __attribute__((ext_vector_type(4))) float v4f;
typedef __attribute__((ext_vector_type(8))) float v8f;

#define EMBED  2048
#define NEXP   64
#define BATCH  16384
#define TOPK   2
#define KSPLIT 8
#define KSLICE (EMBED / KSPLIT)   // 256 K per wave

// ---------------------------------------------------------------------------
// Fused MoE router: one 256-thread block (8 waves) owns a 16-row M-tile.
// Phase 1: each wave computes the full 16x64 partial logits over a 256-wide
//          K-slice with V_WMMA_F32_16X16X4_F32 (full fp32 precision).
// Phase 2: partials -> LDS ([n][m] layout), block reduction + bias.
// Phase 3: 16 threads do per-row top-2 + 2-way softmax from LDS and write
//          gates / indices directly. No logits round-trip through HBM.
// ---------------------------------------------------------------------------
__global__ __launch_bounds__(256)
void moe_router_fused(const float* __restrict__ x,
                      const float* __restrict__ W,
                      const float* __restrict__ bias,
                      float* __restrict__ gates,
                      int* __restrict__ idx_out)
{
    __shared__ float part[KSPLIT * NEXP * 16]; // [kslice][n][m], 32 KB
    __shared__ float lgt[NEXP * 16];           // [n][m], 4 KB

    const int tid  = threadIdx.x;
    const int lane = tid & 31;
    const int wave = tid >> 5;                 // K-split id, 0..7
    const int row0 = blockIdx.x * 16;          // 16-row M-tile

    const int half  = lane >> 4;               // selects K pair in A/B layout
    const int l16   = lane & 15;
    const int kbase = wave * KSLICE;

    // A (16x4 f32): lane -> M=l16, K = k + 2*half + {0,1}  => float2 load
    const float* xrow = x + (size_t)(row0 + l16) * EMBED + kbase + 2 * half;
    // B (4x16 f32): lane -> N=l16, same K pair; B[k][n] = W[n*EMBED + k]
    const float* wr0 = W + (size_t)(l16 +  0) * EMBED + kbase + 2 * half;
    const float* wr1 = W + (size_t)(l16 + 16) * EMBED + kbase + 2 * half;
    const float* wr2 = W + (size_t)(l16 + 32) * EMBED + kbase + 2 * half;
    const float* wr3 = W + (size_t)(l16 + 48) * EMBED + kbase + 2 * half;

    v8f acc0 = {}, acc1 = {}, acc2 = {}, acc3 = {};

    #pragma unroll 4
    for (int k = 0; k < KSLICE; k += 4) {
        v2f a  = *(const v2f*)(xrow + k);
        v2f b0 = *(const v2f*)(wr0  + k);
        v2f b1 = *(const v2f*)(wr1  + k);
        v2f b2 = *(const v2f*)(wr2  + k);
        v2f b3 = *(const v2f*)(wr3  + k);
        acc0 = __builtin_amdgcn_wmma_f32_16x16x4_f32(false, a, false, b0,
                                                     (short)0, acc0, false, false);
        acc1 = __builtin_amdgcn_wmma_f32_16x16x4_f32(false, a, false, b1,
                                                     (short)0, acc1, false, false);
        acc2 = __builtin_amdgcn_wmma_f32_16x16x4_f32(false, a, false, b2,
                                                     (short)0, acc2, false, false);
        acc3 = __builtin_amdgcn_wmma_f32_16x16x4_f32(false, a, false, b3,
                                                     (short)0, acc3, false, false);
    }

    // ---- Phase 2a: store partials to LDS, [n][m] layout (m contiguous) ----
    // C/D layout: VGPR v -> M = v + 8*half, N = l16 (+16*tile)
    {
        float* p = part + wave * (NEXP * 16);
        const int mb = 8 * half;
        *(v4f*)&p[(l16 +  0) * 16 + mb]     = (v4f){acc0[0], acc0[1], acc0[2], acc0[3]};
        *(v4f*)&p[(l16 +  0) * 16 + mb + 4] = (v4f){acc0[4], acc0[5], acc0[6], acc0[7]};
        *(v4f*)&p[(l16 + 16) * 16 + mb]     = (v4f){acc1[0], acc1[1], acc1[2], acc1[3]};
        *(v4f*)&p[(l16 + 16) * 16 + mb + 4] = (v4f){acc1[4], acc1[5], acc1[6], acc1[7]};
        *(v4f*)&p[(l16 + 32) * 16 + mb]     = (v4f){acc2[0], acc2[1], acc2[2], acc2[3]};
        *(v4f*)&p[(l16 + 32) * 16 + mb + 4] = (v4f){acc2[4], acc2[5], acc2[6], acc2[7]};
        *(v4f*)&p[(l16 + 48) * 16 + mb]     = (v4f){acc3[0], acc3[1], acc3[2], acc3[3]};
        *(v4f*)&p[(l16 + 48) * 16 + mb + 4] = (v4f){acc3[4], acc3[5], acc3[6], acc3[7]};
    }
    __syncthreads();

    // ---- Phase 2b: reduce 8 partials; each thread owns 4 consecutive m ----
    {
        const int e0 = tid * 4;          // element in 16x64 tile, [n][m] layout
        const int n  = e0 >> 4;          // all 4 elements share this n
        v4f s = {};
        #pragma unroll
        for (int w = 0; w < KSPLIT; ++w)
            s += *(const v4f*)&part[w * (NEXP * 16) + e0];
        const float bb = bias[n];
        s += (v4f){bb, bb, bb, bb};
        *(v4f*)&lgt[e0] = s;
    }
    __syncthreads();

    // ---- Phase 3: top-2 + softmax, one thread per row ----
    if (tid < 16) {
        const int m = tid;
        float v1 = -INFINITY, v2 = -INFINITY;
        int   i1 = 0,          i2 = 0;
        #pragma unroll 8
        for (int n = 0; n < NEXP; ++n) {
            const float v = lgt[n * 16 + m];
            if (v > v1)      { v2 = v1; i2 = i1; v1 = v; i1 = n; }
            else if (v > v2) { v2 = v;  i2 = n; }
        }
        const float e  = __expf(v2 - v1);       // v1 >= v2: numerically safe
        const float g1 = 1.0f / (1.0f + e);
        const float g2 = 1.0f - g1;

        const int row = row0 + m;
        gates[row * TOPK + 0]   = g1;
        gates[row * TOPK + 1]   = g2;
        idx_out[row * TOPK + 0] = i1;
        idx_out[row * TOPK + 1] = i2;
    }
}

// ---------------------------------------------------------------------------
extern "C" void kernel_launch(void* const* d_in, const int* in_sizes, int n_in,
                              void* d_out, int out_size, void* d_ws, size_t ws_size,
                              hipStream_t stream)
{
    const float* x    = (const float*)d_in[0];   // [BATCH, EMBED]
    const float* W    = (const float*)d_in[1];   // [NEXP, EMBED]
    const float* bias = (const float*)d_in[2];   // [NEXP]

    float* gates = (float*)d_out;                       // BATCH*TOPK floats
    int*   idx   = (int*)d_out + (size_t)BATCH * TOPK;  // BATCH*TOPK int32

    // One block per 16-row tile: 1024 blocks x 256 threads = 8192 waves.
    moe_router_fused<<<dim3(BATCH / 16), dim3(256), 0, stream>>>(x, W, bias, gates, idx);
}